// LGSSM_63823214018790
// MI455X (gfx1250) — compile-verified
//
#include <hip/hip_runtime.h>
#include <hip/hip_bf16.h>

// ---------------------------------------------------------------------------
// LGSSM Kalman filter + RTS smoother for MI455X (gfx1250, wave32).
// One workgroup per batch element; state lives in LDS across all T steps.
// All small GEMMs (32x32, 16x32, 32x16) run on V_WMMA_F32_16X16X4_F32 so the
// covariance algebra (incl. 16x16 / 32x32 inverses) stays full fp32.
// Serial-chain optimizations: exit-barrier-only GEMM helper (half the
// s_barrier traffic), global_prefetch of next-step inputs, NT stores for the
// write-once smoothed outputs and NT loads for the read-once filtered state.
// ---------------------------------------------------------------------------

#define B_   128
#define T_   512
#define Z_   32
#define AD_  16
#define K_   8
#define NTHR 128          // 4 wave32 per block
#define NWAVES 4
#define AUGLD 66          // augmented [M | I] leading dim (max 32x64, padded)
#define OUTSTRIDE 2112    // 2*(Z + Z*Z)
// output slot offsets within one (b,t) record
#define OFF_FM 0
#define OFF_FC 32
#define OFF_SM 1056
#define OFF_SC 1088

typedef __attribute__((ext_vector_type(2))) float v2f;
typedef __attribute__((ext_vector_type(8))) float v8f;

// D[MxN](ldd) = A[MxK](lda) * B   (B is [KxN](ldb), or [NxK](ldb) if transB)
// Each wave computes 16x16 tiles via V_WMMA_F32_16X16X4_F32 (K-steps of 4).
// CONTRACT: caller guarantees inputs are published (a barrier has executed
// after the last write to A/B). This helper only issues the EXIT barrier.
// Tile loop is wave-uniform so EXEC stays all-ones for WMMA.
__device__ __forceinline__ void mm_tiles(float* __restrict__ D, int ldd,
                                         const float* __restrict__ A, int lda,
                                         const float* __restrict__ B, int ldb,
                                         int M, int N, int K, bool transB,
                                         int tid) {
  const int wave = tid >> 5;
  const int lane = tid & 31;
  const int half = lane >> 4;   // 0: lanes 0-15, 1: lanes 16-31
  const int mr   = lane & 15;   // row (A) / col (B,D) index within tile
  const int tilesN = N >> 4;
  const int ntiles = (M >> 4) * tilesN;
  for (int tile = wave; tile < ntiles; tile += NWAVES) {
    const int tm = (tile / tilesN) << 4;
    const int tn = (tile % tilesN) << 4;
    v8f acc = {};
    for (int k0 = 0; k0 < K; k0 += 4) {
      const int ka = k0 + 2 * half;   // ISA f32 A layout: v0/v1 <-> K pairs per lane-half
      v2f a, b;
      a.x = A[(tm + mr) * lda + ka];
      a.y = A[(tm + mr) * lda + ka + 1];
      if (!transB) {
        b.x = B[ka * ldb + tn + mr];
        b.y = B[(ka + 1) * ldb + tn + mr];
      } else {  // B stored [NxK]: B^T on the fly
        b.x = B[(tn + mr) * ldb + ka];
        b.y = B[(tn + mr) * ldb + ka + 1];
      }
      acc = __builtin_amdgcn_wmma_f32_16x16x4_f32(
          /*neg_a=*/false, a, /*neg_b=*/false, b,
          /*c_mod=*/(short)0, acc, /*reuse_a=*/false, /*reuse_b=*/false);
    }
#pragma unroll
    for (int i = 0; i < 8; ++i)   // D layout: VGPR i -> rows i / i+8 by lane half
      D[(tm + i + 8 * half) * ldd + tn + mr] = acc[i];
  }
  __syncthreads();
}

// y[i] = sum_j Mtx[i*ldm+j]*x[j] (+ badd[i]) for i < Mrows (VALU, 1 row/thread)
__device__ __forceinline__ void matvec(float* __restrict__ y,
                                       const float* __restrict__ Mtx, int ldm,
                                       const float* __restrict__ x,
                                       const float* __restrict__ badd,
                                       int Mrows, int Kcols, int tid) {
  __syncthreads();
  if (tid < Mrows) {
    float s = badd ? badd[tid] : 0.0f;
    for (int j = 0; j < Kcols; ++j) s += Mtx[tid * ldm + j] * x[j];
    y[tid] = s;
  }
  __syncthreads();
}

// W[e] = sum_k al[k] * Base[k*nelem + e]   (mixture of K_ base matrices)
__device__ __forceinline__ void mix(float* __restrict__ W,
                                    const float* __restrict__ Base, int nelem,
                                    const float* __restrict__ al, int tid) {
  for (int e = tid; e < nelem; e += NTHR) {
    float s = 0.0f;
#pragma unroll
    for (int k = 0; k < K_; ++k) s += al[k] * Base[k * nelem + e];
    W[e] = s;
  }
}

// Gauss-Jordan inverse of SPD matrix in Aug left half (n x n, ld AUGLD);
// right half [n,2n) must hold identity on entry; inverse ends at Aug + n.
// Pivot = diagonal (no row swaps; valid for SPD covariance matrices).
// First in-loop barrier also publishes the caller's Aug initialization.
__device__ __forceinline__ void gj_invert(float* __restrict__ Aug, int n,
                                          float* __restrict__ fcol, int tid) {
  const int w = 2 * n;
  for (int i = 0; i < n; ++i) {
    __syncthreads();
    if (tid < n) fcol[tid] = Aug[tid * AUGLD + i];   // capture column i
    __syncthreads();
    const float pinv = 1.0f / fcol[i];
    for (int c = tid; c < w; c += NTHR) Aug[i * AUGLD + c] *= pinv;  // normalize row i
    __syncthreads();
    for (int e = tid; e < n * w; e += NTHR) {        // eliminate
      const int r = e / w, c = e - r * w;
      if (r != i) Aug[r * AUGLD + c] -= fcol[r] * Aug[i * AUGLD + c];
    }
  }
  __syncthreads();
}

__global__ __launch_bounds__(NTHR) void lgssm_kernel(
    const float* __restrict__ a_obs,   // [B,T,AD]
    const float* __restrict__ alpha,   // [B,T,K]
    const float* __restrict__ u,       // [B,T,Z]
    const float* __restrict__ Amat,    // [K,Z,Z]
    const float* __restrict__ Bmat,    // [K,Z,Z]
    const float* __restrict__ Cmat,    // [K,AD,Z]
    const float* __restrict__ log_Q,   // [Z]
    const float* __restrict__ log_R,   // [AD]
    const float* __restrict__ z0_mean, // [Z]
    const float* __restrict__ z0_logv, // [Z]
    float* __restrict__ out)           // [B,T,2112]
{
  __shared__ float sA[Z_ * Z_], sBm[Z_ * Z_], sC[AD_ * Z_];
  __shared__ float sZC[Z_ * Z_];                 // zc (filter) / Ps (smoother)
  __shared__ float sZM[Z_];                      // zm (filter) / zs (smoother)
  __shared__ float sT1[Z_ * Z_], sT2[Z_ * Z_], sJ[Z_ * Z_];
  __shared__ float sPp[Z_ * Z_], sD[Z_ * Z_], sFC[Z_ * Z_];
  __shared__ float sAug[Z_ * AUGLD];
  __shared__ float sFM[Z_], sVecA[Z_], sVecB[Z_], sVecC[Z_];
  __shared__ float sAl[K_], sU[Z_], expQ[Z_], expR[AD_], fcol[Z_];

  const int b = blockIdx.x;
  const int tid = threadIdx.x;

  // ---- init: exp(logQ/R), z0 state ----
  if (tid < Z_)  { expQ[tid] = __expf(log_Q[tid]); sZM[tid] = z0_mean[tid]; }
  if (tid < AD_) { expR[tid] = __expf(log_R[tid]); }
  for (int e = tid; e < Z_ * Z_; e += NTHR) {
    const int r = e >> 5, c = e & 31;
    sZC[e] = (r == c) ? __expf(z0_logv[r]) : 0.0f;
  }
  __syncthreads();

  // ================= forward Kalman filter =================
  for (int t = 0; t < T_; ++t) {
    const size_t bt = (size_t)b * T_ + t;
    if (tid < K_)  sAl[tid]  = alpha[bt * K_ + tid];
    if (tid < Z_)  sU[tid]   = u[bt * Z_ + tid];
    if (tid < AD_) sVecC[tid] = a_obs[bt * AD_ + tid];   // observation a_t
    if (t + 1 < T_ && tid == 0) {                        // warm next step's inputs
      __builtin_prefetch(&alpha[(bt + 1) * K_], 0, 3);
      __builtin_prefetch(&u[(bt + 1) * Z_], 0, 3);
      __builtin_prefetch(&a_obs[(bt + 1) * AD_], 0, 3);
    }
    __syncthreads();
    mix(sC, Cmat, AD_ * Z_, sAl, tid);
    if (t > 0) { mix(sA, Amat, Z_ * Z_, sAl, tid); mix(sBm, Bmat, Z_ * Z_, sAl, tid); }
    __syncthreads();

    if (t > 0) {  // ---- predict ----
      matvec(sVecA, sBm, Z_, sU, nullptr, Z_, Z_, tid);         // bu = Bmix @ u
      matvec(sVecB, sA, Z_, sZM, sVecA, Z_, Z_, tid);           // zm' = A zm + bu
      if (tid < Z_) sZM[tid] = sVecB[tid];                      // (next read barriered)
      mm_tiles(sT1, Z_, sA, Z_, sZC, Z_, Z_, Z_, Z_, false, tid);   // T1 = A zc
      mm_tiles(sZC, Z_, sT1, Z_, sA, Z_, Z_, Z_, Z_, true, tid);    // zc = T1 A^T
      if (tid < Z_) sZC[tid * Z_ + tid] += expQ[tid];               // + Q
      __syncthreads();
    }

    // ---- measurement update ----
    matvec(sVecA, sC, Z_, sZM, nullptr, AD_, Z_, tid);            // apred = C zm
    if (tid < AD_) sVecA[tid] = sVecC[tid] - sVecA[tid];          // innov (same owner)
    mm_tiles(sT2, Z_, sC, Z_, sZC, Z_, AD_, Z_, Z_, false, tid);  // T2 = C zc [16x32]
    mm_tiles(sAug, AUGLD, sT2, Z_, sC, Z_, AD_, AD_, Z_, true, tid); // S = T2 C^T
    if (tid < AD_) sAug[tid * AUGLD + tid] += expR[tid];          // + R
    for (int e = tid; e < AD_ * AD_; e += NTHR) {                 // [S | I]
      const int r = e >> 4, c = e & 15;
      sAug[r * AUGLD + AD_ + c] = (r == c) ? 1.0f : 0.0f;
    }
    gj_invert(sAug, AD_, fcol, tid);                              // Sinv @ sAug+16
    mm_tiles(sJ, AD_, sZC, Z_, sC, Z_, Z_, AD_, Z_, true, tid);   // ZCt = zc C^T [32x16]
    mm_tiles(sD, AD_, sJ, AD_, sAug + AD_, AUGLD, Z_, AD_, AD_, false, tid); // Kg
    matvec(sVecB, sD, AD_, sVecA, sZM, Z_, AD_, tid);             // zm += Kg innov
    if (tid < Z_) sZM[tid] = sVecB[tid];                          // (barriered later)
    mm_tiles(sPp, Z_, sD, AD_, sT2, Z_, Z_, Z_, AD_, false, tid); // Kg T2 [32x32]
    for (int e = tid; e < Z_ * Z_; e += NTHR) sZC[e] -= sPp[e];   // zc -= Kg C zc
    __syncthreads();

    // ---- store filtered results (RT: re-read by the smoother, keep in L2) ----
    float* ob = out + bt * OUTSTRIDE;
    if (tid < Z_) ob[OFF_FM + tid] = sZM[tid];
    for (int e = tid; e < Z_ * Z_; e += NTHR) ob[OFF_FC + e] = sZC[e];
    __syncthreads();
  }

  // make filter-phase global writes visible before smoother re-reads them
  __threadfence();
  __syncthreads();

  // ================= backward RTS smoother =================
  {  // t = T-1: smoothed == filtered (current LDS state); write-once -> NT
    float* ob = out + ((size_t)b * T_ + (T_ - 1)) * OUTSTRIDE;
    if (tid < Z_) __builtin_nontemporal_store(sZM[tid], &ob[OFF_SM + tid]);
    for (int e = tid; e < Z_ * Z_; e += NTHR)
      __builtin_nontemporal_store(sZC[e], &ob[OFF_SC + e]);
  }
  for (int t = T_ - 2; t >= 0; --t) {
    const size_t btn = (size_t)b * T_ + (t + 1);   // dynamics at t+1
    const size_t bt  = (size_t)b * T_ + t;
    __syncthreads();
    if (tid < K_) sAl[tid] = alpha[btn * K_ + tid];
    if (tid < Z_) sU[tid]  = u[btn * Z_ + tid];
    const float* obf = out + bt * OUTSTRIDE;       // filtered[t] (written above)
    if (tid < Z_) sFM[tid] = __builtin_nontemporal_load(&obf[OFF_FM + tid]);
    for (int e = tid; e < Z_ * Z_; e += NTHR)
      sFC[e] = __builtin_nontemporal_load(&obf[OFF_FC + e]);
    if (t > 0 && tid < 33) {                       // prefetch prev record (33 lines)
      __builtin_prefetch(out + (bt - 1) * OUTSTRIDE + (size_t)tid * 32, 0, 3);
    }
    __syncthreads();
    mix(sA, Amat, Z_ * Z_, sAl, tid);
    mix(sBm, Bmat, Z_ * Z_, sAl, tid);
    __syncthreads();
    matvec(sVecA, sBm, Z_, sU, nullptr, Z_, Z_, tid);             // bu
    matvec(sVecB, sA, Z_, sFM, sVecA, Z_, Z_, tid);               // zpred
    mm_tiles(sT1, Z_, sA, Z_, sFC, Z_, Z_, Z_, Z_, false, tid);   // A fc
    mm_tiles(sPp, Z_, sT1, Z_, sA, Z_, Z_, Z_, Z_, true, tid);    // Ppred = A fc A^T
    if (tid < Z_) sPp[tid * Z_ + tid] += expQ[tid];               // + Q
    __syncthreads();
    for (int e = tid; e < Z_ * Z_; e += NTHR) {                   // [Ppred | I]
      const int r = e >> 5, c = e & 31;
      sAug[r * AUGLD + c] = sPp[e];
      sAug[r * AUGLD + Z_ + c] = (r == c) ? 1.0f : 0.0f;
    }
    gj_invert(sAug, Z_, fcol, tid);                               // Ppinv @ sAug+32
    mm_tiles(sT1, Z_, sFC, Z_, sA, Z_, Z_, Z_, Z_, true, tid);    // fc A^T
    mm_tiles(sJ, Z_, sT1, Z_, sAug + Z_, AUGLD, Z_, Z_, Z_, false, tid); // J
    if (tid < Z_) sVecA[tid] = sZM[tid] - sVecB[tid];             // zs - zpred
    __syncthreads();
    matvec(sVecC, sJ, Z_, sVecA, sFM, Z_, Z_, tid);               // zs' = fm + J dv
    if (tid < Z_) sZM[tid] = sVecC[tid];                          // (barriered later)
    for (int e = tid; e < Z_ * Z_; e += NTHR) sD[e] = sZC[e] - sPp[e]; // Ps - Ppred
    __syncthreads();                                              // publish sD
    mm_tiles(sT2, Z_, sJ, Z_, sD, Z_, Z_, Z_, Z_, false, tid);    // J (Ps-Ppred)
    mm_tiles(sPp, Z_, sT2, Z_, sJ, Z_, Z_, Z_, Z_, true, tid);    // ... J^T
    for (int e = tid; e < Z_ * Z_; e += NTHR) sZC[e] = sFC[e] + sPp[e]; // Ps'
    __syncthreads();
    float* ob = out + bt * OUTSTRIDE;                             // write-once -> NT
    if (tid < Z_) __builtin_nontemporal_store(sZM[tid], &ob[OFF_SM + tid]);
    for (int e = tid; e < Z_ * Z_; e += NTHR)
      __builtin_nontemporal_store(sZC[e], &ob[OFF_SC + e]);
  }
}

extern "C" void kernel_launch(void* const* d_in, const int* in_sizes, int n_in,
                              void* d_out, int out_size, void* d_ws, size_t ws_size,
                              hipStream_t stream) {
  (void)in_sizes; (void)n_in; (void)out_size; (void)d_ws; (void)ws_size;
  const float* a_obs  = (const float*)d_in[0];
  const float* alpha  = (const float*)d_in[1];
  const float* u      = (const float*)d_in[2];
  const float* Amat   = (const float*)d_in[3];
  const float* Bmat   = (const float*)d_in[4];
  const float* Cmat   = (const float*)d_in[5];
  const float* log_Q  = (const float*)d_in[6];
  const float* log_R  = (const float*)d_in[7];
  const float* z0m    = (const float*)d_in[8];
  const float* z0lv   = (const float*)d_in[9];
  float* out = (float*)d_out;
  lgssm_kernel<<<B_, NTHR, 0, stream>>>(a_obs, alpha, u, Amat, Bmat, Cmat,
                                        log_Q, log_R, z0m, z0lv, out);
}